// GATMechanism_46110768890276
// MI455X (gfx1250) — compile-verified
//
#include <hip/hip_runtime.h>
#include <hip/hip_bf16.h>
#include <math.h>

#define NN 2048
#define D_IN 128
#define D_OUT 64

typedef __attribute__((ext_vector_type(2))) float v2f;
typedef __attribute__((ext_vector_type(4))) float v4f;
typedef __attribute__((ext_vector_type(8))) float v8f;

// ---------------------------------------------------------------------------
// Kernel 1: h[2048,64] = X[2048,128] @ W[64,128]^T  via V_WMMA_F32_16X16X4_F32
// One wave per 16x16 output tile; 128 M-tiles x 4 N-tiles = 512 tiles.
// Fragment layouts per CDNA5 ISA 7.12.2 (32-bit A 16x4, 32-bit C/D 16x16).
// ---------------------------------------------------------------------------
__global__ __launch_bounds__(256) void gat_linear_wmma(
    const float* __restrict__ X, const float* __restrict__ W,
    float* __restrict__ h) {
  const int wave  = threadIdx.x >> 5;
  const int lane  = threadIdx.x & 31;
  const int tile  = blockIdx.x * 8 + wave;     // 0..511
  const int tM    = tile >> 2;                 // 0..127
  const int tN    = tile & 3;                  // 0..3
  const int mBase = tM * 16, nBase = tN * 16;
  const int lrow  = lane & 15;                 // M (for A) / N (for B)
  const int khalf = lane >> 4;                 // 0: K={k,k+1}, 1: K={k+2,k+3}

  const float* __restrict__ arow = X + (size_t)(mBase + lrow) * D_IN;
  const float* __restrict__ brow = W + (size_t)(nBase + lrow) * D_IN; // B[k][n]=W[n][k]

  v8f c = {};
#pragma unroll
  for (int k = 0; k < D_IN; k += 4) {
    v2f a = *(const v2f*)(arow + k + 2 * khalf);   // 8-byte aligned (k even)
    v2f b = *(const v2f*)(brow + k + 2 * khalf);
    // 8 args: (neg_a, A, neg_b, B, c_mod, C, reuse_a, reuse_b)
    c = __builtin_amdgcn_wmma_f32_16x16x4_f32(false, a, false, b,
                                              (short)0, c, false, false);
  }
#pragma unroll
  for (int r = 0; r < 8; ++r) {
    const int row = mBase + khalf * 8 + r;         // C/D: M = 8*(lane>=16)+vgpr
    h[(size_t)row * D_OUT + nBase + lrow] = c[r];  // N = lane&15
  }
}

// ---------------------------------------------------------------------------
// Kernel 2: s_i = h @ a[:64], s_j = h @ a[64:]   (one thread per row)
// ---------------------------------------------------------------------------
__global__ void gat_scores(const float* __restrict__ h,
                           const float* __restrict__ a,
                           float* __restrict__ s_i, float* __restrict__ s_j) {
  const int i = blockIdx.x * blockDim.x + threadIdx.x;
  if (i >= NN) return;
  const float* hr = h + (size_t)i * D_OUT;
  float si = 0.f, sj = 0.f;
#pragma unroll
  for (int c = 0; c < D_OUT; ++c) {
    si += hr[c] * a[c];
    sj += hr[c] * a[D_OUT + c];
  }
  s_i[i] = si;
  s_j[i] = sj;
}

// ---------------------------------------------------------------------------
// Kernel 3: one block per row i. Build masked leaky-ReLU scores on the fly,
// row softmax (max + sum reductions, exps cached in LDS), then stream
// messages[i,j,:] = alpha[i,j]*h[j,:] with coalesced non-temporal 128-bit
// stores (this 1 GiB write stream is the roofline limiter: ~46 us at 23.3TB/s).
// ---------------------------------------------------------------------------
__global__ __launch_bounds__(256) void gat_softmax_messages(
    const float* __restrict__ h, const unsigned char* __restrict__ mask,
    const float* __restrict__ s_i, const float* __restrict__ s_j,
    float* __restrict__ out) {
  __shared__ float alpha[NN];   // 8 KB: unnormalized exps
  __shared__ float red[256];

  const int i = blockIdx.x;
  const int t = threadIdx.x;
  const float si = s_i[i];
  const unsigned char* __restrict__ mrow = mask + (size_t)i * NN;

  // Pass 1: masked row max of leaky_relu(s_i + s_j)
  float m = -INFINITY;
  for (int j = t; j < NN; j += 256) {
    if (mrow[j]) {
      float s = si + s_j[j];
      s = s > 0.f ? s : 0.01f * s;
      m = fmaxf(m, s);
    }
  }
  red[t] = m;
  __syncthreads();
  for (int off = 128; off > 0; off >>= 1) {
    if (t < off) red[t] = fmaxf(red[t], red[t + off]);
    __syncthreads();
  }
  const float rowmax = red[0];
  __syncthreads();

  // Pass 2: exps + row sum
  float acc = 0.f;
  for (int j = t; j < NN; j += 256) {
    float e = 0.f;
    if (mrow[j]) {
      float s = si + s_j[j];
      s = s > 0.f ? s : 0.01f * s;
      e = __expf(s - rowmax);
      acc += e;
    }
    alpha[j] = e;
  }
  red[t] = acc;
  __syncthreads();
  for (int off = 128; off > 0; off >>= 1) {
    if (t < off) red[t] += red[t + off];
    __syncthreads();
  }
  const float sum = red[0];
  const float inv = sum > 0.f ? 1.0f / sum : 0.f;  // invalid rows -> all zeros
  __syncthreads();

  // Pass 3: stream the output row. 16 threads per j, 128 bits each -> each
  // iteration writes 16 j's x 256 B = 4 KB fully coalesced, non-temporal.
  const int jLocal = t >> 4;   // 0..15
  const int c4     = t & 15;   // 0..15 (v4f index within 64 floats)
  const v4f* __restrict__ h4 = (const v4f*)h;
  for (int jb = 0; jb < NN; jb += 16) {
    const int j = jb + jLocal;
    const float aj = alpha[j] * inv;
    const v4f hv = h4[(size_t)j * 16 + c4];
    v4f o = aj * hv;
    v4f* dst = (v4f*)(out + ((size_t)i * NN + j) * D_OUT) + c4;
    __builtin_nontemporal_store(o, dst);
  }
}

// ---------------------------------------------------------------------------
extern "C" void kernel_launch(void* const* d_in, const int* in_sizes, int n_in,
                              void* d_out, int out_size, void* d_ws, size_t ws_size,
                              hipStream_t stream) {
  const float*         X    = (const float*)d_in[0];          // [2048,128]
  const unsigned char* mask = (const unsigned char*)d_in[1];  // [2048,2048] bool
  const float*         W    = (const float*)d_in[2];          // [64,128]
  const float*         a    = (const float*)d_in[3];          // [128]
  float*               out  = (float*)d_out;                  // [2048,2048,64]

  float* h   = (float*)d_ws;              // 2048*64 floats = 512 KB
  float* s_i = h + (size_t)NN * D_OUT;    // 2048 floats
  float* s_j = s_i + NN;                  // 2048 floats

  gat_linear_wmma<<<64, 256, 0, stream>>>(X, W, h);
  gat_scores<<<(NN + 255) / 256, 256, 0, stream>>>(h, a, s_i, s_j);
  gat_softmax_messages<<<NN, 256, 0, stream>>>(h, mask, s_i, s_j, out);
}